// Att_Net_18004502905269
// MI455X (gfx1250) — compile-verified
//
#include <hip/hip_runtime.h>
#include <math.h>

typedef __attribute__((ext_vector_type(2))) float v2f;
typedef __attribute__((ext_vector_type(8))) float v8f;

#define CDIM 128
#define INV_SQRT_C 0.08838834764831845f  // 1/sqrt(128)

// ---- sortable-uint encoding for float atomic max ----
__device__ __forceinline__ unsigned f2sort(float f) {
    unsigned u = __float_as_uint(f);
    return (u & 0x80000000u) ? ~u : (u | 0x80000000u);
}
__device__ __forceinline__ float sort2f(unsigned u) {
    return (u & 0x80000000u) ? __uint_as_float(u & 0x7FFFFFFFu)
                             : __uint_as_float(~u);
}

// ================= Projection GEMMs: out = x @ W + b (fp32 WMMA) ============
// One wave per 16x16 output tile. K-loop in steps of 4 -> v_wmma_f32_16x16x4_f32.
__global__ __launch_bounds__(128) void proj_wmma(
    const float* __restrict__ x,
    const float* __restrict__ Wq, const float* __restrict__ bq,
    const float* __restrict__ Wk, const float* __restrict__ bk,
    const float* __restrict__ Wv, const float* __restrict__ bv,
    const float* __restrict__ Ws, const float* __restrict__ bs,
    float* __restrict__ Q, float* __restrict__ K,
    float* __restrict__ V, float* __restrict__ S,
    int nMTiles)
{
    const int wave = blockIdx.x * (blockDim.x >> 5) + (threadIdx.x >> 5);
    const int lane = threadIdx.x & 31;
    const int tilesPerMat = nMTiles * (CDIM / 16);
    const int mat = wave / tilesPerMat;
    if (mat >= 4) return;
    const int rem   = wave % tilesPerMat;
    const int mtile = rem >> 3;      // node tile
    const int ntile = rem & 7;       // output-column tile

    const float* W; const float* bias; float* O;
    if      (mat == 0) { W = Wq; bias = bq; O = Q; }
    else if (mat == 1) { W = Wk; bias = bk; O = K; }
    else if (mat == 2) { W = Wv; bias = bv; O = V; }
    else               { W = Ws; bias = bs; O = S; }

    const int lrow = lane & 15;           // row within tile (A) / col within tile (B,D)
    const int kb   = (lane >> 4) << 1;    // K sub-pair base: 0 or 2
    const int row  = mtile * 16 + lrow;   // global node row for A
    const int colg = ntile * 16 + lrow;   // global output column for B/D

    // C init = bias broadcast down the column
    v8f c;
    {
        const float bv0 = bias[colg];
#pragma unroll
        for (int i = 0; i < 8; ++i) c[i] = bv0;
    }

    const float* xrow = x + (size_t)row * CDIM + kb;   // (kk+kb) even -> 8B aligned
#pragma unroll
    for (int kk = 0; kk < CDIM; kk += 4) {
        v2f a = *(const v2f*)(xrow + kk);              // A: x[row, kk+kb .. +1]
        v2f bb;
        bb.x = W[(size_t)(kk + kb)     * CDIM + colg]; // B: W[kk+kb,   col]
        bb.y = W[(size_t)(kk + kb + 1) * CDIM + colg]; // B: W[kk+kb+1, col]
        c = __builtin_amdgcn_wmma_f32_16x16x4_f32(
                /*neg_a=*/false, a, /*neg_b=*/false, bb,
                /*c_mod=*/(short)0, c, /*reuse_a=*/false, /*reuse_b=*/false);
    }

    // D layout: VGPR i -> M = i + (lane>=16 ? 8 : 0), N = lane&15
    const int orow = mtile * 16 + ((lane >> 4) << 3);
#pragma unroll
    for (int i = 0; i < 8; ++i)
        O[(size_t)(orow + i) * CDIM + colg] = c[i];
}

// ============ Edge phase 1: alpha_raw = dot(q[dst],k[src])/sqrt(C); segment max over src
__global__ __launch_bounds__(256) void edge_dot(
    const float* __restrict__ Q, const float* __restrict__ K,
    const long long* __restrict__ idx, float* __restrict__ alpha,
    unsigned* __restrict__ amax, int E)
{
    const int e    = blockIdx.x * 8 + (threadIdx.x >> 5);
    const int lane = threadIdx.x & 31;
    if (e >= E) return;
    const long long s = idx[e];       // source j
    const long long d = idx[E + e];   // target i
    const float4 qv = ((const float4*)(Q + (size_t)d * CDIM))[lane];
    const float4 kv = ((const float4*)(K + (size_t)s * CDIM))[lane];
    float p = qv.x * kv.x + qv.y * kv.y + qv.z * kv.z + qv.w * kv.w;
#pragma unroll
    for (int off = 16; off > 0; off >>= 1) p += __shfl_xor(p, off, 32);
    if (lane == 0) {
        const float a = p * INV_SQRT_C;
        alpha[e] = a;
        atomicMax(&amax[s], f2sort(a));
    }
}

// ============ Edge phase 2: alpha = exp(a - amax[src]); denom[src] += alpha
__global__ __launch_bounds__(256) void edge_exp(
    const long long* __restrict__ idx, float* __restrict__ alpha,
    const unsigned* __restrict__ amax, float* __restrict__ denom, int E)
{
    const int e = blockIdx.x * 256 + threadIdx.x;
    if (e >= E) return;
    const long long s = idx[e];
    const float m  = sort2f(amax[s]);
    const float ex = expf(alpha[e] - m);
    alpha[e] = ex;
    atomicAdd(&denom[s], ex);
}

// ============ Edge phase 3: agg[dst] += v[src] * alpha/(denom[src]+eps)
__global__ __launch_bounds__(256) void edge_scatter(
    const float* __restrict__ V, const long long* __restrict__ idx,
    const float* __restrict__ alpha, const float* __restrict__ denom,
    float* __restrict__ agg, int E)
{
    const int e    = blockIdx.x * 8 + (threadIdx.x >> 5);
    const int lane = threadIdx.x & 31;
    if (e >= E) return;
    const long long s = idx[e];
    const long long d = idx[E + e];
    const float w = alpha[e] / (denom[s] + 1e-16f);
    const float4 vv = ((const float4*)(V + (size_t)s * CDIM))[lane];
    float* o = agg + (size_t)d * CDIM + lane * 4;
    atomicAdd(o + 0, vv.x * w);
    atomicAdd(o + 1, vv.y * w);
    atomicAdd(o + 2, vv.z * w);
    atomicAdd(o + 3, vv.w * w);
}

// ============ Final: out = ELU( LN(agg + skip) * g + b ), one wave per node
__global__ __launch_bounds__(256) void ln_elu(
    const float* __restrict__ agg, const float* __restrict__ S,
    const float* __restrict__ g, const float* __restrict__ bta,
    float* __restrict__ out, int N)
{
    const int n    = blockIdx.x * 8 + (threadIdx.x >> 5);
    const int lane = threadIdx.x & 31;
    if (n >= N) return;
    float4 o  = ((const float4*)(agg + (size_t)n * CDIM))[lane];
    float4 sk = ((const float4*)(S   + (size_t)n * CDIM))[lane];
    o.x += sk.x; o.y += sk.y; o.z += sk.z; o.w += sk.w;

    float sum = o.x + o.y + o.z + o.w;
#pragma unroll
    for (int off = 16; off > 0; off >>= 1) sum += __shfl_xor(sum, off, 32);
    const float mu = sum * (1.0f / CDIM);

    const float dx = o.x - mu, dy = o.y - mu, dz = o.z - mu, dw = o.w - mu;
    float sq = dx * dx + dy * dy + dz * dz + dw * dw;
#pragma unroll
    for (int off = 16; off > 0; off >>= 1) sq += __shfl_xor(sq, off, 32);
    const float rstd = rsqrtf(sq * (1.0f / CDIM) + 1e-5f);

    const float4 gg = ((const float4*)g)[lane];
    const float4 bb = ((const float4*)bta)[lane];
    float4 h;
    h.x = dx * rstd * gg.x + bb.x;
    h.y = dy * rstd * gg.y + bb.y;
    h.z = dz * rstd * gg.z + bb.z;
    h.w = dw * rstd * gg.w + bb.w;
    h.x = (h.x > 0.0f) ? h.x : expm1f(h.x);
    h.y = (h.y > 0.0f) ? h.y : expm1f(h.y);
    h.z = (h.z > 0.0f) ? h.z : expm1f(h.z);
    h.w = (h.w > 0.0f) ? h.w : expm1f(h.w);
    ((float4*)(out + (size_t)n * CDIM))[lane] = h;
}

extern "C" void kernel_launch(void* const* d_in, const int* in_sizes, int n_in,
                              void* d_out, int out_size, void* d_ws, size_t ws_size,
                              hipStream_t stream) {
    const float*     x   = (const float*)d_in[0];
    const long long* idx = (const long long*)d_in[1];
    const float* Wq = (const float*)d_in[2];  const float* bq = (const float*)d_in[3];
    const float* Wk = (const float*)d_in[4];  const float* bk = (const float*)d_in[5];
    const float* Wv = (const float*)d_in[6];  const float* bv = (const float*)d_in[7];
    const float* Ws = (const float*)d_in[8];  const float* bs = (const float*)d_in[9];
    const float* lg = (const float*)d_in[10]; const float* lb = (const float*)d_in[11];
    float* out = (float*)d_out;

    const int N = in_sizes[0] / CDIM;   // 20000
    const int E = in_sizes[1] / 2;      // 320000
    const size_t NC = (size_t)N * CDIM;

    // workspace carve-up
    float*    Q     = (float*)d_ws;
    float*    K     = Q + NC;
    float*    V     = K + NC;
    float*    S     = V + NC;
    float*    AGG   = S + NC;
    float*    ALPHA = AGG + NC;
    unsigned* AMAX  = (unsigned*)(ALPHA + E);
    float*    DEN   = (float*)(AMAX + N);

    // zero AGG, ALPHA, AMAX (0 == sortable -inf floor), DEN in one memset
    hipMemsetAsync(AGG, 0, (NC + (size_t)E + 2 * (size_t)N) * sizeof(float), stream);

    const int nMTiles    = N / 16;                   // 1250
    const int totalWaves = nMTiles * (CDIM / 16) * 4; // 40000 tiles
    proj_wmma<<<(totalWaves + 3) / 4, 128, 0, stream>>>(
        x, Wq, bq, Wk, bk, Wv, bv, Ws, bs, Q, K, V, S, nMTiles);

    edge_dot<<<(E + 7) / 8, 256, 0, stream>>>(Q, K, idx, ALPHA, AMAX, E);
    edge_exp<<<(E + 255) / 256, 256, 0, stream>>>(idx, ALPHA, AMAX, DEN, E);
    edge_scatter<<<(E + 7) / 8, 256, 0, stream>>>(V, idx, ALPHA, DEN, AGG, E);
    ln_elu<<<(N + 7) / 8, 256, 0, stream>>>(AGG, S, lg, lb, out, N);
}